// GCNNet_70480413327361
// MI455X (gfx1250) — compile-verified
//
#include <hip/hip_runtime.h>
#include <hip/hip_bf16.h>

#define NN      100000
#define NE      1600000
#define HID     128
#define NLAYERS 4
#define NCLS    6
#define EPSV    1e-5f

typedef float v2f __attribute__((ext_vector_type(2)));
typedef float v8f __attribute__((ext_vector_type(8)));

// ---------------- utility ----------------
__global__ void zero_kernel(float* __restrict__ p, int n) {
    int i = blockIdx.x * blockDim.x + threadIdx.x;
    if (i < n) p[i] = 0.0f;
}

// ---------------- degrees (float atomics -> global_atomic_add_f32) ----------------
__global__ void degree_kernel(const int* __restrict__ src, const int* __restrict__ dst,
                              float* __restrict__ deg_out, float* __restrict__ deg_in, int nE) {
    int e = blockIdx.x * blockDim.x + threadIdx.x;
    if (e < nE) {
        unsafeAtomicAdd(&deg_out[src[e]], 1.0f);
        unsafeAtomicAdd(&deg_in[dst[e]], 1.0f);
    }
}

__global__ void norm_kernel(const float* __restrict__ deg_out, const float* __restrict__ deg_in,
                            float* __restrict__ norm_src, float* __restrict__ norm_dst, int n) {
    int i = blockIdx.x * blockDim.x + threadIdx.x;
    if (i < n) {
        float a = deg_out[i];
        float b = deg_in[i];
        norm_src[i] = (a > 0.0f) ? rsqrtf(fmaxf(a, 1.0f)) : 0.0f;
        norm_dst[i] = (b > 0.0f) ? rsqrtf(fmaxf(b, 1.0f)) : 0.0f;
    }
}

// ---------------- embedding gather ----------------
__global__ void embed_kernel(const int* __restrict__ h, const float* __restrict__ emb,
                             float* __restrict__ x, int n) {
    int i = blockIdx.x * blockDim.x + threadIdx.x;
    if (i < n * HID) {
        int r = i >> 7;        // / HID
        int j = i & (HID - 1); // % HID
        x[i] = emb[h[r] * HID + j];
    }
}

// ---------------- edge scatter: agg[dst] += x[src] * norm_src[src] ----------------
// One wave (32 lanes) per edge; each lane moves a float4 (128 floats / edge).
// x (51 MB) is L2-resident on MI455X (192 MB L2), so this is an L2-bandwidth pass.
__global__ void scatter_kernel(const int* __restrict__ src, const int* __restrict__ dst,
                               const float* __restrict__ x, const float* __restrict__ norm_src,
                               float* __restrict__ agg, int nE) {
    int gid  = blockIdx.x * blockDim.x + threadIdx.x;
    int edge = gid >> 5;
    int lane = threadIdx.x & 31;
    if (edge < nE) {
        int s = src[edge];
        int d = dst[edge];
        float ns = norm_src[s];
        float4 v = ((const float4*)(x + (size_t)s * HID))[lane];
        float* ad = agg + (size_t)d * HID + lane * 4;
        unsafeAtomicAdd(ad + 0, v.x * ns);
        unsafeAtomicAdd(ad + 1, v.y * ns);
        unsafeAtomicAdd(ad + 2, v.z * ns);
        unsafeAtomicAdd(ad + 3, v.w * ns);
    }
}

// ---------------- WMMA fp32 GEMM: out = op(rowScale.*A) @ W + bias ----------------
// A: [M x K] row-major (row optionally scaled by rowScale[row]); W: [K x N] row-major.
// One wave per 16x16 output tile; K stepped by 4 with V_WMMA_F32_16X16X4_F32.
// A-frag 16x4 (ISA 7.12.2): v0 = K={0,2}, v1 = K={1,3}, split by lane-half.
// B-frag 4x16: rows striped across lanes within a VGPR, same half-split.
__global__ void gemm_wmma_kernel(const float* __restrict__ A, const float* __restrict__ rowScale,
                                 const float* __restrict__ W, const float* __restrict__ bias,
                                 float* __restrict__ out, int M, int K, int N, int doRelu) {
    int tileM = blockIdx.x;
    int tileN = blockIdx.y;
    int lane  = threadIdx.x & 31;
    int half  = lane >> 4;   // 0 or 1
    int lid   = lane & 15;
    int m0 = tileM * 16;
    int n0 = tileN * 16;
    int row = m0 + lid;      // both lane-halves carry the same 16 A-rows

    float rs = rowScale ? rowScale[row] : 1.0f;
    const float* arow = A + (size_t)row * K;

    v8f acc = {};
    for (int k = 0; k < K; k += 4) {
        int ka = k + 2 * half;
        v2f a, b;
        a.x = arow[ka]     * rs;                // A[row][ka]
        a.y = arow[ka + 1] * rs;                // A[row][ka+1]
        b.x = W[(size_t)ka       * N + n0 + lid]; // W[ka][col]
        b.y = W[(size_t)(ka + 1) * N + n0 + lid]; // W[ka+1][col]
        acc = __builtin_amdgcn_wmma_f32_16x16x4_f32(
            /*neg_a=*/false, a, /*neg_b=*/false, b,
            /*c_mod=*/(short)0, acc, /*reuse_a=*/false, /*reuse_b=*/false);
    }

    float bv = bias[n0 + lid];
#pragma unroll
    for (int v = 0; v < 8; ++v) {
        int r = m0 + v + 8 * half;              // D layout: VGPR v -> rows {v, v+8}
        float val = acc[v] + bv;
        if (doRelu) val = fmaxf(val, 0.0f);
        out[(size_t)r * N + n0 + lid] = val;
    }
}

// ---------------- BatchNorm stats: per-column sum / sumsq ----------------
#define BN_ROWS 256
__global__ void bn_stats_kernel(const float* __restrict__ xh,
                                float* __restrict__ sums, float* __restrict__ sumsq, int M) {
    int j  = threadIdx.x;            // 0..127 (column)
    int r0 = blockIdx.x * BN_ROWS;
    int r1 = r0 + BN_ROWS; if (r1 > M) r1 = M;
    float s = 0.0f, s2 = 0.0f;
    for (int r = r0; r < r1; ++r) {
        float v = xh[(size_t)r * HID + j];
        s  += v;
        s2 += v * v;
    }
    unsafeAtomicAdd(&sums[j], s);
    unsafeAtomicAdd(&sumsq[j], s2);
}

__global__ void bn_final_kernel(const float* __restrict__ sums, const float* __restrict__ sumsq,
                                const float* __restrict__ gamma, const float* __restrict__ beta,
                                float* __restrict__ aMul, float* __restrict__ aAdd, int M) {
    int j = threadIdx.x;             // 0..127
    float invM = 1.0f / (float)M;
    float mu   = sums[j] * invM;
    float var  = sumsq[j] * invM - mu * mu;
    float inv  = rsqrtf(var + EPSV);
    float g    = gamma[j] * inv;
    aMul[j] = g;
    aAdd[j] = beta[j] - mu * g;
}

// ---------------- normalize + ReLU + residual (in-place on x) ----------------
__global__ void bn_apply_kernel(const float* __restrict__ xh, const float* __restrict__ aMul,
                                const float* __restrict__ aAdd, float* __restrict__ x, int M) {
    int i = blockIdx.x * blockDim.x + threadIdx.x;
    if (i < M * HID) {
        int j = i & (HID - 1);
        float v = fmaxf(xh[i] * aMul[j] + aAdd[j], 0.0f);
        x[i] = v + x[i];
    }
}

// ---------------- readout tail: [M x 32] @ [32 x 6] + b3 ----------------
__global__ void readout_kernel(const float* __restrict__ y2, const float* __restrict__ W3,
                               const float* __restrict__ b3, float* __restrict__ out, int M) {
    int r = blockIdx.x * blockDim.x + threadIdx.x;
    if (r < M) {
        float acc[NCLS];
#pragma unroll
        for (int c = 0; c < NCLS; ++c) acc[c] = b3[c];
        const float* yr = y2 + (size_t)r * 32;
#pragma unroll
        for (int k = 0; k < 32; ++k) {
            float v = yr[k];
#pragma unroll
            for (int c = 0; c < NCLS; ++c) acc[c] += v * W3[k * NCLS + c];
        }
#pragma unroll
        for (int c = 0; c < NCLS; ++c) out[(size_t)r * NCLS + c] = acc[c];
    }
}

// ---------------- host side ----------------
extern "C" void kernel_launch(void* const* d_in, const int* in_sizes, int n_in,
                              void* d_out, int out_size, void* d_ws, size_t ws_size,
                              hipStream_t stream) {
    const int*   h     = (const int*)  d_in[0];
    const int*   src   = (const int*)  d_in[1];
    const int*   dst   = (const int*)  d_in[2];
    const float* emb   = (const float*)d_in[3];
    const float* Wl    = (const float*)d_in[4];   // [4,128,128]
    const float* bl    = (const float*)d_in[5];   // [4,128]
    const float* gamma = (const float*)d_in[6];   // [4,128]
    const float* beta  = (const float*)d_in[7];   // [4,128]
    const float* W1    = (const float*)d_in[8];   // [128,64]
    const float* b1    = (const float*)d_in[9];   // [64]
    const float* W2    = (const float*)d_in[10];  // [64,32]
    const float* b2    = (const float*)d_in[11];  // [32]
    const float* W3    = (const float*)d_in[12];  // [32,6]
    const float* b3    = (const float*)d_in[13];  // [6]
    float* out = (float*)d_out;

    // workspace carve-out (256B aligned slabs)
    char*  ws  = (char*)d_ws;
    size_t off = 0;
    auto alloc = [&](size_t nfloats) -> float* {
        float* p = (float*)(ws + off);
        off += ((nfloats * sizeof(float)) + 255) & ~(size_t)255;
        return p;
    };
    float* x        = alloc((size_t)NN * HID);  // node features (residual stream)
    float* agg      = alloc((size_t)NN * HID);  // scatter target; reused as y1
    float* xh       = alloc((size_t)NN * HID);  // GEMM output; reused as y2
    float* deg_out  = alloc(NN);
    float* deg_in   = alloc(NN);
    float* norm_src = alloc(NN);
    float* norm_dst = alloc(NN);
    float* sums     = alloc(HID);
    float* sumsq    = alloc(HID);
    float* aMul     = alloc(HID);
    float* aAdd     = alloc(HID);
    (void)ws_size; (void)in_sizes; (void)n_in; (void)out_size;

    const int TB = 256;
    const int nElem  = NN * HID;                 // 12.8M
    const int gElem  = (nElem + TB - 1) / TB;
    const int gNode  = (NN + TB - 1) / TB;
    const int gEdge  = (NE + TB - 1) / TB;
    const int gScat  = (NE * 32 + TB - 1) / TB;  // one wave per edge

    // degrees + norms
    zero_kernel<<<gNode, TB, 0, stream>>>(deg_out, NN);
    zero_kernel<<<gNode, TB, 0, stream>>>(deg_in, NN);
    degree_kernel<<<gEdge, TB, 0, stream>>>(src, dst, deg_out, deg_in, NE);
    norm_kernel<<<gNode, TB, 0, stream>>>(deg_out, deg_in, norm_src, norm_dst, NN);

    // x = emb[h]
    embed_kernel<<<gElem, TB, 0, stream>>>(h, emb, x, NN);

    // GCN layers
    dim3 gemmGrid(NN / 16, HID / 16);            // 6250 x 8, exact (100000 = 16*6250)
    for (int l = 0; l < NLAYERS; ++l) {
        zero_kernel<<<gElem, TB, 0, stream>>>(agg, nElem);
        scatter_kernel<<<gScat, TB, 0, stream>>>(src, dst, x, norm_src, agg, NE);
        gemm_wmma_kernel<<<gemmGrid, 32, 0, stream>>>(
            agg, norm_dst, Wl + (size_t)l * HID * HID, bl + (size_t)l * HID,
            xh, NN, HID, HID, /*doRelu=*/0);
        zero_kernel<<<1, TB, 0, stream>>>(sums, HID);
        zero_kernel<<<1, TB, 0, stream>>>(sumsq, HID);
        bn_stats_kernel<<<(NN + BN_ROWS - 1) / BN_ROWS, HID, 0, stream>>>(xh, sums, sumsq, NN);
        bn_final_kernel<<<1, HID, 0, stream>>>(sums, sumsq,
                                               gamma + (size_t)l * HID, beta + (size_t)l * HID,
                                               aMul, aAdd, NN);
        bn_apply_kernel<<<gElem, TB, 0, stream>>>(xh, aMul, aAdd, x, NN);
    }

    // readout MLP: 128 -> 64 -> 32 -> 6
    float* y1 = agg;  // [NN x 64]
    float* y2 = xh;   // [NN x 32]
    dim3 g1(NN / 16, 64 / 16);
    gemm_wmma_kernel<<<g1, 32, 0, stream>>>(x,  nullptr, W1, b1, y1, NN, 128, 64, /*doRelu=*/1);
    dim3 g2(NN / 16, 32 / 16);
    gemm_wmma_kernel<<<g2, 32, 0, stream>>>(y1, nullptr, W2, b2, y2, NN, 64, 32, /*doRelu=*/1);
    readout_kernel<<<gNode, TB, 0, stream>>>(y2, W3, b3, out, NN);
}